// CausalGNN_49211735277597
// MI455X (gfx1250) — compile-verified
//
#include <hip/hip_runtime.h>
#include <hip/hip_bf16.h>
#include <math.h>

typedef __attribute__((ext_vector_type(2))) float v2f;
typedef __attribute__((ext_vector_type(8))) float v8f;

#define B_GRAPH 4096
#define N_SUB   200000
#define DIM     256
#define HID     512
#define T_OUT   10

// workspace layout (float offsets)
#define WS_SUMV 0ul            // [4096*256] -> becomes h_sub_aligned in place
#define WS_SUME 1048576ul      // [4096*256] -> becomes h_sub_env in place
#define WS_CNTV 2097152ul      // [4096]
#define WS_CNTE 2101248ul      // [4096]
#define WS_AN   2105344ul      // [4096]
#define WS_EN   2109440ul      // [4096]
#define WS_NZP  2113536ul      // [4096]
#define WS_NZE  2117632ul      // [4096]
#define WS_SA   2121728ul      // [256]
#define WS_SE   2121984ul      // [256]
#define WS_SCAL 2122240ul      // [8]: 0=nz_pos count, 1=nz_env count
#define WS_TOTAL 2122248ul

// ---------------------------------------------------------------- zero init
__global__ void k_zero(float* __restrict__ ws, float* __restrict__ loss_slot) {
  size_t i = (size_t)blockIdx.x * blockDim.x + threadIdx.x;
  size_t stride = (size_t)gridDim.x * blockDim.x;
  for (size_t p = i; p < WS_TOTAL; p += stride) ws[p] = 0.f;
  if (i == 0) *loss_slot = 0.f;
}

// ------------------------------------------- fused mask MLP (WMMA) + scatter
// block = 128 threads (4 waves), 64 rows of h_sub per block (16 per wave)
__global__ __launch_bounds__(128)
void k_mlp_scatter(const float* __restrict__ h_sub,
                   const float* __restrict__ W1,
                   const float* __restrict__ b1,
                   const float* __restrict__ W2,
                   const float* __restrict__ b2,
                   const int* __restrict__ sub2graph,
                   float* __restrict__ ws,
                   float* __restrict__ mask_out)
{
  constexpr int A_STR = 260;                 // padded row stride (dwords)
  constexpr int AW    = 16 * A_STR;          // 4160 per-wave A tile
  constexpr int AOFF  = 0;
  constexpr int WSOFF = 4 * AW;              // 16640: W1 slab (transposed)
  constexpr int W_STR = 34;                  // padded col stride (dwords)
  constexpr int SRED  = WSOFF + 256 * W_STR; // 25344: 4*32*8 reduce scratch
  constexpr int FLG   = SRED + 1024;         // 26368: 4*16*2 flags
  __shared__ float lds[26496];

  const int tid    = threadIdx.x;
  const int w      = tid >> 5;
  const int lane   = tid & 31;
  const int laneLo = lane & 15;
  const int hi     = lane >> 4;
  const int row0   = blockIdx.x * 64;

  // stage 64x256 A tiles into LDS (coalesced float4 reads)
  #pragma unroll
  for (int i = 0; i < 32; ++i) {
    int e = (tid + i * 128) * 4;             // 16384 floats
    int r = e >> 8;
    int c = e & 255;
    const float4 v = *(const float4*)(h_sub + (size_t)(row0 + r) * DIM + c);
    *(float4*)(&lds[AOFF + (r >> 4) * AW + (r & 15) * A_STR + c]) = v;
  }
  __syncthreads();

  const int aBase = AOFF + w * AW + laneLo * A_STR + 2 * hi;
  float p[8];
  #pragma unroll
  for (int j = 0; j < 8; ++j) p[j] = 0.f;

  for (int chunk = 0; chunk < 2; ++chunk) {
    const int cbase = chunk * 256;
    v8f acc[16] = {};
    for (int slab = 0; slab < 8; ++slab) {
      const int k0 = slab * 32;
      __syncthreads();
      // cooperatively load + transpose W1[k0:k0+32][cbase:cbase+256] into LDS
      #pragma unroll
      for (int i = 0; i < 16; ++i) {
        int e = (tid + i * 128) * 4;         // 8192 floats
        int r = e >> 8;                      // 0..31 (k within slab)
        int c = e & 255;                     // col within chunk
        const float4 v = *(const float4*)(W1 + (size_t)(k0 + r) * HID + cbase + c);
        lds[WSOFF + (c + 0) * W_STR + r] = v.x;
        lds[WSOFF + (c + 1) * W_STR + r] = v.y;
        lds[WSOFF + (c + 2) * W_STR + r] = v.z;
        lds[WSOFF + (c + 3) * W_STR + r] = v.w;
      }
      __syncthreads();
      #pragma unroll
      for (int ks = 0; ks < 8; ++ks) {
        const v2f a = *(const v2f*)(&lds[aBase + k0 + ks * 4]);
        #pragma unroll
        for (int t = 0; t < 16; ++t) {
          const v2f bb = *(const v2f*)(&lds[WSOFF + (t * 16 + laneLo) * W_STR + ks * 4 + 2 * hi]);
          acc[t] = __builtin_amdgcn_wmma_f32_16x16x4_f32(
              false, a, false, bb, (short)0, acc[t], false, false);
        }
      }
    }
    // fused ReLU + b1 + dot with W2 (per-lane partials)
    #pragma unroll
    for (int t = 0; t < 16; ++t) {
      const int col = cbase + t * 16 + laneLo;
      const float b1v = b1[col];
      const float w2v = W2[col];
      #pragma unroll
      for (int j = 0; j < 8; ++j) {
        float h = acc[t][j] + b1v;
        h = h > 0.f ? h : 0.f;
        p[j] += h * w2v;
      }
    }
  }

  // cross-lane reduce p[j] -> per-row score s (C layout: row = j + 8*hi)
  #pragma unroll
  for (int j = 0; j < 8; ++j) lds[SRED + w * 256 + lane * 8 + j] = p[j];
  __syncthreads();
  const float b2v = b2[0];
  if (lane < 16) {
    const int r = lane;
    const int j = r & 7;
    const int lb = (r < 8) ? 0 : 16;
    float s = b2v;
    #pragma unroll
    for (int l = 0; l < 16; ++l) s += lds[SRED + w * 256 + (lb + l) * 8 + j];
    const float m = 1.f / (1.f + expf(-s));
    mask_out[row0 + w * 16 + r] = m;
    lds[FLG + w * 32 + r * 2 + 0] = (m > 0.4f) ? 1.f : 0.f;
    lds[FLG + w * 32 + r * 2 + 1] = (m <= 0.3f) ? 1.f : 0.f;
  }
  __syncthreads();

  // scatter masked rows into per-graph sums (L2-resident f32 atomics)
  float* sumv = ws + WS_SUMV;
  float* sume = ws + WS_SUME;
  float* cntv = ws + WS_CNTV;
  float* cnte = ws + WS_CNTE;
  for (int r = 0; r < 16; ++r) {
    const float fv = lds[FLG + w * 32 + r * 2 + 0];
    const float fe = lds[FLG + w * 32 + r * 2 + 1];
    if (fv == 0.f && fe == 0.f) continue;
    const int g = sub2graph[row0 + w * 16 + r];
    const int ab = AOFF + w * AW + r * A_STR;
    #pragma unroll
    for (int i = 0; i < 8; ++i) {
      const int c = lane + i * 32;
      const float val = lds[ab + c];
      if (fv != 0.f) atomicAdd(&sumv[(size_t)g * DIM + c], val);
      if (fe != 0.f) atomicAdd(&sume[(size_t)g * DIM + c], val);
    }
    if (lane == 0) {
      if (fv != 0.f) atomicAdd(&cntv[g], 1.f);
      if (fe != 0.f) atomicAdd(&cnte[g], 1.f);
    }
  }
}

// -------------------- per-graph mean, norms, nz flags, global S_a/S_e sums
__global__ __launch_bounds__(256)
void k_seg_finalize(float* __restrict__ ws)
{
  __shared__ float red[1024];
  __shared__ float bc2[2];
  const int i = blockIdx.x;
  const int tid = threadIdx.x;
  float* sv = ws + WS_SUMV + (size_t)i * DIM;
  float* se = ws + WS_SUME + (size_t)i * DIM;
  const float cv = fmaxf(ws[WS_CNTV + i], 1.f);
  const float ce = fmaxf(ws[WS_CNTE + i], 1.f);
  const float a = sv[tid] / cv;
  const float e = se[tid] / ce;
  sv[tid] = a;                               // h_sub_aligned in place
  se[tid] = e;                               // h_sub_env in place
  red[tid]       = a * a;
  red[256 + tid] = e * e;
  red[512 + tid] = (a != 0.f) ? 1.f : 0.f;
  red[768 + tid] = (e != 0.f) ? 1.f : 0.f;
  __syncthreads();
  for (int s = 128; s > 0; s >>= 1) {
    if (tid < s) {
      red[tid]       += red[tid + s];
      red[256 + tid] += red[256 + tid + s];
      red[512 + tid] += red[512 + tid + s];
      red[768 + tid] += red[768 + tid + s];
    }
    __syncthreads();
  }
  if (tid == 0) {
    const float an = fmaxf(sqrtf(red[0]),   1e-8f);
    const float en = fmaxf(sqrtf(red[256]), 1e-8f);
    const float nzp = (red[512] > 0.f) ? 1.f : 0.f;
    const float nze = (red[768] > 0.f) ? 1.f : 0.f;
    ws[WS_AN + i] = an;  ws[WS_EN + i] = en;
    ws[WS_NZP + i] = nzp; ws[WS_NZE + i] = nze;
    atomicAdd(&ws[WS_SCAL + 0], nzp);
    atomicAdd(&ws[WS_SCAL + 1], nze);
    bc2[0] = an; bc2[1] = en;
  }
  __syncthreads();
  // S_a = sum_j a_j/an_j ; S_e = sum_j nz_j e_j/en_j (zero rows contribute 0)
  atomicAdd(&ws[WS_SA + tid], a / bc2[0]);
  atomicAdd(&ws[WS_SE + tid], e / bc2[1]);
}

// ---------------- contrastive loss via collapsed row-sums of cosine matrices
__global__ __launch_bounds__(256)
void k_loss(const float* __restrict__ ws, float* __restrict__ loss)
{
  __shared__ float red[512];
  const int i = blockIdx.x;
  const int tid = threadIdx.x;
  const float a = ws[WS_SUMV + (size_t)i * DIM + tid];
  red[tid]       = a * ws[WS_SA + tid];
  red[256 + tid] = a * ws[WS_SE + tid];
  __syncthreads();
  for (int s = 128; s > 0; s >>= 1) {
    if (tid < s) { red[tid] += red[tid + s]; red[256 + tid] += red[256 + tid + s]; }
    __syncthreads();
  }
  if (tid == 0) {
    const float an = ws[WS_AN + i];
    const float nzp = ws[WS_NZP + i];
    const float nzpCnt = ws[WS_SCAL + 0];
    const float nzeCnt = ws[WS_SCAL + 1];
    const float pos_num = fmaxf(nzpCnt - 1.f, 1.f);
    const float positive = ((float)B_GRAPH - red[0] / an) / pos_num;
    const float negative = (nzeCnt - red[256] / an) / fmaxf(nzeCnt, 1.f);
    float contrib = positive - negative + 1.f;
    contrib = contrib > 0.f ? contrib : 0.f;
    contrib *= nzp * ((nzeCnt > 0.f) ? 1.f : 0.f);
    atomicAdd(loss, contrib * (1.f / (float)B_GRAPH));
  }
}

// ------------------------------- classifier head: concat(h_graph,aligned)@Wc
__global__ __launch_bounds__(256)
void k_out(const float* __restrict__ h_graph,
           const float* __restrict__ ws,
           const float* __restrict__ Wc,
           const float* __restrict__ bcv,
           float* __restrict__ out)
{
  __shared__ float red[T_OUT * 256];
  const int i = blockIdx.x;
  const int tid = threadIdx.x;
  const float x0 = h_graph[(size_t)i * DIM + tid];
  const float x1 = ws[WS_SUMV + (size_t)i * DIM + tid];
  #pragma unroll
  for (int t = 0; t < T_OUT; ++t) {
    red[t * 256 + tid] = x0 * Wc[(size_t)tid * T_OUT + t]
                       + x1 * Wc[(size_t)(DIM + tid) * T_OUT + t];
  }
  __syncthreads();
  if (tid < T_OUT) {
    float s = bcv[tid];
    for (int k = 0; k < 256; ++k) s += red[tid * 256 + k];
    out[(size_t)i * T_OUT + tid] = s;
  }
}

extern "C" void kernel_launch(void* const* d_in, const int* in_sizes, int n_in,
                              void* d_out, int out_size, void* d_ws, size_t ws_size,
                              hipStream_t stream) {
  const float* h_graph = (const float*)d_in[0];
  const float* h_sub   = (const float*)d_in[1];
  const float* W1      = (const float*)d_in[2];
  const float* b1      = (const float*)d_in[3];
  const float* W2      = (const float*)d_in[4];
  const float* b2      = (const float*)d_in[5];
  const float* Wc      = (const float*)d_in[6];
  const float* bc      = (const float*)d_in[7];
  const int*   s2g     = (const int*)d_in[8];
  float* out  = (float*)d_out;                      // [4096,10]
  float* loss = out + (size_t)B_GRAPH * T_OUT;      // scalar
  float* mask = loss + 1;                           // [200000]
  float* ws   = (float*)d_ws;

  k_zero<<<2048, 256, 0, stream>>>(ws, loss);
  k_mlp_scatter<<<N_SUB / 64, 128, 0, stream>>>(h_sub, W1, b1, W2, b2, s2g, ws, mask);
  k_seg_finalize<<<B_GRAPH, 256, 0, stream>>>(ws);
  k_loss<<<B_GRAPH, 256, 0, stream>>>(ws, loss);
  k_out<<<B_GRAPH, 256, 0, stream>>>(h_graph, ws, Wc, bc, out);
}